// Transformer_attn_80951543595334
// MI455X (gfx1250) — compile-verified
//
#include <hip/hip_runtime.h>
#include <hip/hip_bf16.h>
#include <math.h>

// CDNA5 / gfx1250. wave32. Uses V_WMMA_F32_16X16X4_F32 (exact fp32 matrix core).
typedef float v2f __attribute__((ext_vector_type(2)));
typedef float v8f __attribute__((ext_vector_type(8)));

#define CC 192          // channels
#define BB 32           // batch
#define MASKED 512      // masked pixels per patch (32 rows x 16 alternating cols)
#define JAC_SCALE 98304.0f   // p * (p/2) * C = 32*16*192

// ---------------------------------------------------------------------------
// Kernel A: M = Wq1^T Wk1 + Wq2^T Wk2 + Wq3^T Wk3   (192x192, row-major [c][c'])
// One wave per 16x16 tile; 12x12 tiles; K=192 in steps of 4 per product.
// ---------------------------------------------------------------------------
__global__ __launch_bounds__(32) void mgram_kernel(
    const float* __restrict__ wq1, const float* __restrict__ wk1,
    const float* __restrict__ wq2, const float* __restrict__ wk2,
    const float* __restrict__ wq3, const float* __restrict__ wk3,
    float* __restrict__ Mws)
{
    const int lane = threadIdx.x;
    const int ct   = (blockIdx.x % 12) * 16;   // row tile (c)
    const int ctp  = (blockIdx.x / 12) * 16;   // col tile (c')
    const int half = lane >> 4;                // 0: lanes 0-15, 1: lanes 16-31
    const int l16  = lane & 15;

    const float* WQ[3] = { wq1, wq2, wq3 };
    const float* WK[3] = { wk1, wk2, wk3 };

    v8f acc{};
    for (int p = 0; p < 3; ++p) {
        const float* wq = WQ[p];
        const float* wk = WK[p];
        #pragma unroll 4
        for (int k0 = 0; k0 < CC; k0 += 4) {
            const int ka = k0 + half * 2;      // this half-wave's K rows (k, k+1)
            v2f a, b;
            // A[i,k] = Wq[o=k0+k][ct+i]  (i = l16)
            a.x = wq[(ka + 0) * CC + ct + l16];
            a.y = wq[(ka + 1) * CC + ct + l16];
            // B[k,j] = Wk[o=k0+k][ctp+j] (j = l16)
            b.x = wk[(ka + 0) * CC + ctp + l16];
            b.y = wk[(ka + 1) * CC + ctp + l16];
            acc = __builtin_amdgcn_wmma_f32_16x16x4_f32(
                false, a, false, b, (short)0, acc, false, false);
        }
    }
    // D layout: VGPR rr -> row (rr + 8*half), col = ctp + l16
    const int row0 = ct + half * 8;
    #pragma unroll
    for (int rr = 0; rr < 8; ++rr)
        Mws[(row0 + rr) * CC + ctp + l16] = acc[rr];
}

// ---------------------------------------------------------------------------
// Kernel B: raw logits for the two same-parity 2x2 attention blocks.
//   logits[n,m] = sum_{x masked} X_n[:,x]^T M X_m[:,x]
// One workgroup per (b, parity): patches n0=parity (top half), n1=parity+2
// (bottom half). Masked X chunks (192 x 64 pixels, both patches) staged in LDS
// (96 KB). Y = M^T X_n computed 16x16-tile-wise with WMMA, immediately dotted
// against X_{n0}, X_{n1} tiles from LDS; reduced with ds_add_f32.
// ---------------------------------------------------------------------------
__global__ __launch_bounds__(256) void logits_kernel(
    const float* __restrict__ xg, const float* __restrict__ Mws,
    float* __restrict__ logits_ws)
{
    __shared__ float Xs[2][CC][64];   // [patch-in-pair][c][chunk pixel]  (96 KB)
    __shared__ float sAcc[4];

    const int tid  = threadIdx.x;
    const int bp   = blockIdx.x;      // 0..63
    const int b    = bp >> 1;
    const int par  = bp & 1;          // parity == w_idx of the patch pair
    const int widx = par;

    if (tid < 4) sAcc[tid] = 0.0f;

    const int lane = tid & 31, wave = tid >> 5;
    const int half = lane >> 4, l16 = lane & 15;

    float pAcc[4] = {0.f, 0.f, 0.f, 0.f};   // [nn*2 + m]

    for (int chunk = 0; chunk < 8; ++chunk) {
        __syncthreads();              // also orders sAcc init / prev compute
        // Stage masked pixels [chunk*64, chunk*64+64) for both patches.
        for (int t = tid; t < 2 * CC * 64; t += 256) {
            const int nn   = t / (CC * 64);
            const int rem  = t - nn * CC * 64;
            const int c    = rem >> 6;
            const int jloc = rem & 63;
            const int j    = chunk * 64 + jloc;     // masked pixel index 0..511
            const int py   = j >> 4;
            const int px   = ((j & 15) << 1) + par; // alternating columns
            const int h    = nn * 32 + py;          // nn=0: top, nn=1: bottom
            const int w    = widx * 32 + px;
            Xs[nn][c][jloc] =
                xg[((((size_t)b) * CC + c) * 64 + h) * 64 + w];
        }
        __syncthreads();

        // 96 tiles: nn in {0,1} x 12 c'-tiles x 4 x-tiles; 12 per wave.
        for (int tt = wave; tt < 96; tt += 8) {
            const int nn  = tt / 48;
            const int r   = tt - nn * 48;
            const int cpt = (r >> 2) * 16;   // c' tile base
            const int xt  = (r & 3) * 16;    // pixel tile base within chunk

            v8f acc{};
            #pragma unroll 4
            for (int k0 = 0; k0 < CC; k0 += 4) {
                const int ka = k0 + half * 2;
                v2f a, bf;
                // A[i,k] = M[k0+k][cpt+i]   (Y = M^T X)
                a.x  = Mws[(ka + 0) * CC + cpt + l16];
                a.y  = Mws[(ka + 1) * CC + cpt + l16];
                // B[k,j] = X_n[k0+k][xt+j]
                bf.x = Xs[nn][ka + 0][xt + l16];
                bf.y = Xs[nn][ka + 1][xt + l16];
                acc  = __builtin_amdgcn_wmma_f32_16x16x4_f32(
                    false, a, false, bf, (short)0, acc, false, false);
            }
            // Frobenius dot of Y-tile with X_m tiles (m = 0,1).
            const int jj = l16;
            const int ib = half * 8;
            #pragma unroll
            for (int m = 0; m < 2; ++m) {
                float s = 0.0f;
                #pragma unroll
                for (int rr = 0; rr < 8; ++rr)
                    s += acc[rr] * Xs[m][cpt + ib + rr][xt + jj];
                pAcc[nn * 2 + m] += s;
            }
        }
    }
    __syncthreads();
    #pragma unroll
    for (int q = 0; q < 4; ++q) atomicAdd(&sAcc[q], pAcc[q]);   // ds_add_f32
    __syncthreads();
    if (tid < 4) logits_ws[bp * 4 + tid] = sAcc[tid];
}

// ---------------------------------------------------------------------------
// Kernel C: attn coefficients + logdet. One lane per batch.
// ---------------------------------------------------------------------------
__global__ __launch_bounds__(32) void attn_kernel(
    const float* __restrict__ logits_ws,
    const float* __restrict__ offset,  const float* __restrict__ offset2,
    const float* __restrict__ offset3, const float* __restrict__ scale,
    float* __restrict__ attn_ws, float* __restrict__ logdet_out)
{
    const int b = threadIdx.x;
    if (b >= BB) return;
    const float off  = *offset;
    const float off2 = *offset2;
    const float off3 = *offset3;
    const float sc   = *scale;

    float ld = 0.0f;
    for (int par = 0; par < 2; ++par) {
        float v[2][2];
        for (int i = 0; i < 2; ++i)
            for (int jm = 0; jm < 2; ++jm) {
                const float raw = logits_ws[((b * 2 + par) * 2 + i) * 2 + jm];
                const float l   = raw / sc + off2;
                float val = 1.0f / (1.0f + expf(-l)) + off3;
                if (i == jm) val += off;               // + offset * eye
                v[i][jm] = val;
            }
        const float det = v[0][0] * v[1][1] - v[0][1] * v[1][0];
        ld += logf(fabsf(det));
        for (int i = 0; i < 2; ++i) {
            const int n = par + 2 * i;
            attn_ws[b * 8 + n * 2 + 0] = v[i][i];      // diagonal coef
            attn_ws[b * 8 + n * 2 + 1] = v[i][1 - i];  // partner coef
        }
    }
    logdet_out[b] = ld * JAC_SCALE;
}

// ---------------------------------------------------------------------------
// Kernel D: streaming output. z layout == x layout (unpatchify(patchify)=id).
// (w_idx + px) even -> identity; odd -> 2-term mix with vertical partner.
// float4 along w; each thread handles row h and its partner h+32.
// ---------------------------------------------------------------------------
__global__ __launch_bounds__(256) void out_kernel(
    const float* __restrict__ xg, const float* __restrict__ attn_ws,
    float* __restrict__ zout)
{
    const size_t t = (size_t)blockIdx.x * 256 + threadIdx.x;
    const int wq4 = (int)(t & 15);          // w/4
    const int h0  = (int)((t >> 4) & 31);   // row in top half
    const int bc  = (int)(t >> 9);          // b*192 + c
    const int c   = bc % CC;
    const int b   = bc / CC;
    if (b >= BB) return;

    const int w    = wq4 * 4;
    const int widx = w >> 5;
    const int n_top = widx;        // h_idx = 0
    const int n_bot = 2 + widx;    // h_idx = 1

    const size_t base_top = ((((size_t)b) * CC + c) * 64 + h0) * 64 + w;
    const size_t base_bot = base_top + (size_t)32 * 64;

    const float4 xt4 = *(const float4*)(xg + base_top);
    const float4 xb4 = *(const float4*)(xg + base_bot);
    const float* xt = (const float*)&xt4;
    const float* xb = (const float*)&xb4;

    const float atd = attn_ws[b * 8 + n_top * 2 + 0];
    const float ato = attn_ws[b * 8 + n_top * 2 + 1];
    const float abd = attn_ws[b * 8 + n_bot * 2 + 0];
    const float abo = attn_ws[b * 8 + n_bot * 2 + 1];

    float4 ot4, ob4;
    float* ot = (float*)&ot4;
    float* ob = (float*)&ob4;
    #pragma unroll
    for (int e = 0; e < 4; ++e) {
        if (((widx + e) & 1) == 0) {          // mask==1: identity
            ot[e] = xt[e];
            ob[e] = xb[e];
        } else {                              // mask==0: attention mix
            ot[e] = atd * xt[e] + ato * xb[e];
            ob[e] = abd * xb[e] + abo * xt[e];
        }
    }
    *(float4*)(zout + base_top) = ot4;
    *(float4*)(zout + base_bot) = ob4;
}

// ---------------------------------------------------------------------------
extern "C" void kernel_launch(void* const* d_in, const int* in_sizes, int n_in,
                              void* d_out, int out_size, void* d_ws, size_t ws_size,
                              hipStream_t stream)
{
    (void)in_sizes; (void)n_in; (void)out_size; (void)ws_size;

    const float* x   = (const float*)d_in[0];
    const float* wq1 = (const float*)d_in[1];
    const float* wk1 = (const float*)d_in[2];
    const float* wq2 = (const float*)d_in[3];
    const float* wk2 = (const float*)d_in[4];
    const float* wq3 = (const float*)d_in[5];
    const float* wk3 = (const float*)d_in[6];
    const float* offset  = (const float*)d_in[7];
    const float* offset2 = (const float*)d_in[8];
    const float* offset3 = (const float*)d_in[9];
    const float* scale   = (const float*)d_in[10];

    float* zout       = (float*)d_out;                 // 32*192*64*64 floats
    float* logdet_out = zout + (size_t)32 * 192 * 64 * 64;  // + 32 floats

    float* ws        = (float*)d_ws;
    float* Mws       = ws;            // 192*192      = 36864 floats
    float* logits_ws = ws + 36864;    // 64*4         = 256 floats
    float* attn_ws   = ws + 37120;    // 32*8         = 256 floats

    mgram_kernel<<<144, 32, 0, stream>>>(wq1, wk1, wq2, wk2, wq3, wk3, Mws);
    logits_kernel<<<64, 256, 0, stream>>>(x, Mws, logits_ws);
    attn_kernel<<<1, 32, 0, stream>>>(logits_ws, offset, offset2, offset3,
                                      scale, attn_ws, logdet_out);
    out_kernel<<<12288, 256, 0, stream>>>(x, attn_ws, zout);
}